// GCN_70463233458892
// MI455X (gfx1250) — compile-verified
//
#include <hip/hip_runtime.h>

typedef __attribute__((ext_vector_type(2))) float v2f;
typedef __attribute__((ext_vector_type(8))) float v8f;

#define D_FEAT 64
#define UNITS 64
#define LDS_STRIDE 72   // 64 + 8: makes lanes 0-15 vs 16-31 hit disjoint bank sets

// ---------------------------------------------------------------------------
// Kernel 0: zero the float accumulator (d_ws is poisoned, never re-zeroed)
// ---------------------------------------------------------------------------
__global__ __launch_bounds__(256) void zero_f4(float4* __restrict__ p, int n4) {
  int i = blockIdx.x * blockDim.x + threadIdx.x;
  if (i < n4) p[i] = make_float4(0.f, 0.f, 0.f, 0.f);
}

// ---------------------------------------------------------------------------
// Kernel 1: SpMM scatter. 16 threads per edge, each owns 4 features.
// float4 gather from the embedding row (L2-resident), then 4 native
// global_atomic_add_f32 into the accumulator.
// ---------------------------------------------------------------------------
__global__ __launch_bounds__(256) void spmm_scatter(
    const float* __restrict__ emb, const int* __restrict__ src,
    const int* __restrict__ dst, const float* __restrict__ w,
    float* __restrict__ agg, int nEdges) {
  int gid = blockIdx.x * blockDim.x + threadIdx.x;
  int e = gid >> 4;
  if (e >= nEdges) return;
  int c = (gid & 15) << 2;            // feature chunk base (0,4,...,60)
  int s = src[e];
  int d = dst[e];
  float we = w[e];
  const float4 v =
      *reinterpret_cast<const float4*>(emb + (size_t)s * D_FEAT + c);
  float* o = agg + (size_t)d * D_FEAT + c;
  unsafeAtomicAdd(o + 0, v.x * we);
  unsafeAtomicAdd(o + 1, v.y * we);
  unsafeAtomicAdd(o + 2, v.z * we);
  unsafeAtomicAdd(o + 3, v.w * we);
}

// ---------------------------------------------------------------------------
// Kernel 2: out = relu(agg @ kern) with V_WMMA_F32_16X16X4_F32 (full f32).
// One wave computes a 16x64 output tile: 4 N-tiles x 16 K-steps = 64 WMMAs.
//
// A 16x4 f32 fragment (ISA 7.12.2): M = lane&15; lanes 0-15 carry K={0,1}
// in VGPR0/1, lanes 16-31 carry K={2,3}. B 4x16 mirrors it with N = lane&15.
// C/D 16x16 f32: VGPR r holds M = r + 8*(lane>=16), N = lane&15.
// ---------------------------------------------------------------------------
__global__ __launch_bounds__(256) void wmma_gemm_relu(
    const float* __restrict__ agg, const float* __restrict__ kern,
    float* __restrict__ out, int nRows) {
  __shared__ float sK[D_FEAT * LDS_STRIDE];

  // Cooperative copy of the 64x64 weight into padded LDS (float4 moves).
  for (int i = threadIdx.x; i < (D_FEAT * UNITS) / 4; i += 256) {
    int k = i >> 4;            // 16 float4 per 64-float row
    int n = (i & 15) << 2;
    float4 t = reinterpret_cast<const float4*>(kern)[i];
    *reinterpret_cast<float4*>(&sK[k * LDS_STRIDE + n]) = t;
  }
  __syncthreads();

  const int lane = threadIdx.x & 31;            // wave32
  const int wave = threadIdx.x >> 5;
  const int tile = blockIdx.x * 8 + wave;       // uniform across the wave
  const int nTiles = (nRows + 15) >> 4;
  if (tile >= nTiles) return;                   // wave-uniform: EXEC stays all-1

  const int col = lane & 15;
  const int khalf = (lane >= 16) ? 2 : 0;
  int row = tile * 16 + col;
  if (row >= nRows) row = nRows - 1;            // clamp: duplicate reads only
  const float* __restrict__ arow = agg + (size_t)row * D_FEAT;

  v8f acc0 = {}, acc1 = {}, acc2 = {}, acc3 = {};

#pragma unroll
  for (int kb = 0; kb < D_FEAT / 4; ++kb) {
    const int k0 = kb * 4 + khalf;
    v2f a;
    a.x = arow[k0];
    a.y = arow[k0 + 1];
    const float* b0p = &sK[k0 * LDS_STRIDE + col];
    const float* b1p = &sK[(k0 + 1) * LDS_STRIDE + col];
    v2f b;
    b.x = b0p[0];  b.y = b1p[0];
    acc0 = __builtin_amdgcn_wmma_f32_16x16x4_f32(false, a, false, b,
                                                 (short)0, acc0, false, false);
    b.x = b0p[16]; b.y = b1p[16];
    acc1 = __builtin_amdgcn_wmma_f32_16x16x4_f32(false, a, false, b,
                                                 (short)0, acc1, false, false);
    b.x = b0p[32]; b.y = b1p[32];
    acc2 = __builtin_amdgcn_wmma_f32_16x16x4_f32(false, a, false, b,
                                                 (short)0, acc2, false, false);
    b.x = b0p[48]; b.y = b1p[48];
    acc3 = __builtin_amdgcn_wmma_f32_16x16x4_f32(false, a, false, b,
                                                 (short)0, acc3, false, false);
  }

  // Store with ReLU. Lanes 0-15 write rows mbase..mbase+7, lanes 16-31 +8.
  const int mbase = tile * 16 + ((lane >= 16) ? 8 : 0);
#pragma unroll
  for (int r = 0; r < 8; ++r) {
    int m = mbase + r;
    if (m < nRows) {
      size_t o = (size_t)m * UNITS + col;
      out[o + 0]  = fmaxf(acc0[r], 0.f);
      out[o + 16] = fmaxf(acc1[r], 0.f);
      out[o + 32] = fmaxf(acc2[r], 0.f);
      out[o + 48] = fmaxf(acc3[r], 0.f);
    }
  }
}

// ---------------------------------------------------------------------------
extern "C" void kernel_launch(void* const* d_in, const int* in_sizes, int n_in,
                              void* d_out, int out_size, void* d_ws,
                              size_t ws_size, hipStream_t stream) {
  const float* emb  = (const float*)d_in[0];  // [N, 64]
  const int*   esrc = (const int*)d_in[1];    // [E]
  const int*   edst = (const int*)d_in[2];    // [E]
  const float* ew   = (const float*)d_in[3];  // [E]
  const float* kern = (const float*)d_in[4];  // [64, 64]
  float* out = (float*)d_out;                 // [N, 64]
  float* agg = (float*)d_ws;                  // scratch accumulator [N, 64]

  const int nNodes = in_sizes[0] / D_FEAT;
  const int nEdges = in_sizes[1];

  // Phase 0: zero accumulator
  const int n4 = (nNodes * D_FEAT) / 4;
  zero_f4<<<(n4 + 255) / 256, 256, 0, stream>>>((float4*)agg, n4);

  // Phase 1: edge-parallel scatter-add (16 threads / edge)
  const long long tEdge = (long long)nEdges * 16;
  const int edgeBlocks = (int)((tEdge + 255) / 256);
  spmm_scatter<<<edgeBlocks, 256, 0, stream>>>(emb, esrc, edst, ew, agg,
                                               nEdges);

  // Phase 2: f32 WMMA GEMM + ReLU (8 waves/block, 1 M-tile per wave)
  const int nTiles = (nNodes + 15) / 16;
  const int gemmBlocks = (nTiles + 7) / 8;
  wmma_gemm_relu<<<gemmBlocks, 256, 0, stream>>>(agg, kern, out, nNodes);
}